// MMD_loss_21045339750991
// MI455X (gfx1250) — compile-verified
//
#include <hip/hip_runtime.h>

typedef __attribute__((ext_vector_type(2))) float v2f;
typedef __attribute__((ext_vector_type(8))) float v8f;

#define DIM     256
#define B_HALF  4096
#define N_TOTAL 8192

// ws layout (floats):
// [0]  sumSq accumulator
// [1]  pair-sum accumulator
// [2]  bandwidth
// [16 .. 16+256)   colsum[256] accumulator
// [512 .. 512+8192) sq[row]

__global__ void __launch_bounds__(256) mmd_init_kernel(float* ws) {
    ws[16 + threadIdx.x] = 0.0f;           // colsum
    if (threadIdx.x < 3) ws[threadIdx.x] = 0.0f;
}

// one block (256 threads) per row: sq[row] = sum_d T[row,d]^2 ; atomic sumSq
__global__ void __launch_bounds__(256) mmd_rownorm_kernel(const float* __restrict__ src,
                                                          const float* __restrict__ tgt,
                                                          float* __restrict__ ws) {
    int row = blockIdx.x;
    const float* T = (row < B_HALF) ? (src + (size_t)row * DIM)
                                    : (tgt + (size_t)(row - B_HALF) * DIM);
    float v = T[threadIdx.x];
    __shared__ float red[256];
    red[threadIdx.x] = v * v;
    __syncthreads();
    for (int s = 128; s > 0; s >>= 1) {
        if (threadIdx.x < (unsigned)s) red[threadIdx.x] += red[threadIdx.x + s];
        __syncthreads();
    }
    if (threadIdx.x == 0) {
        ws[512 + row] = red[0];
        atomicAdd(&ws[0], red[0]);
    }
}

// 64 blocks x 256 threads: block b partial-sums columns over rows [b*128, b*128+128)
// (128-row chunks never straddle the src/tgt boundary at 4096 = 32*128)
__global__ void __launch_bounds__(256) mmd_colsum_kernel(const float* __restrict__ src,
                                                         const float* __restrict__ tgt,
                                                         float* __restrict__ ws) {
    const int d  = threadIdx.x;
    const int r0 = blockIdx.x * 128;
    const float* base = (r0 < B_HALF) ? (src + (size_t)r0 * DIM)
                                      : (tgt + (size_t)(r0 - B_HALF) * DIM);
    float s = 0.0f;
    #pragma unroll 8
    for (int r = 0; r < 128; ++r) s += base[(size_t)r * DIM + d];
    atomicAdd(&ws[16 + d], s);
}

// one block: sum(gram) = |colsum|^2 ; bandwidth = (2n*sumSq - 2*sumGram)/(n^2 - n)
__global__ void __launch_bounds__(256) mmd_bw_kernel(float* __restrict__ ws) {
    float s = ws[16 + threadIdx.x];
    __shared__ float red[256];
    red[threadIdx.x] = s * s;
    __syncthreads();
    for (int st = 128; st > 0; st >>= 1) {
        if (threadIdx.x < (unsigned)st) red[threadIdx.x] += red[threadIdx.x + st];
        __syncthreads();
    }
    if (threadIdx.x == 0) {
        float sumSq   = ws[0];
        float sumGram = red[0];
        const float n = (float)N_TOTAL;
        float sumL2 = 2.0f * n * sumSq - 2.0f * sumGram;
        ws[2] = sumL2 / (n * n - n);
    }
}

// per-tile fused epilogue: L2 -> 5-sigma kernel mixture (1 exp + 4 squarings)
// -> symmetry weight (2 above diag / 1 on / 0 below) -> signed accumulation
__device__ __forceinline__ float tile_contrib(const v8f& c, int giBase, int gj,
                                              float sqj, float ej,
                                              const float* __restrict__ sq, float scale) {
    float acc = 0.0f;
    #pragma unroll
    for (int r = 0; r < 8; ++r) {
        const int gi = giBase + r;
        const float sqi = sq[gi];
        const float ei  = (gi < B_HALF) ? 1.0f : -1.0f;
        const float w   = (gj > gi) ? 2.0f : ((gj == gi) ? 1.0f : 0.0f);
        const float L2  = sqi + sqj - 2.0f * c[r];
        const float e   = __expf(-L2 * scale);           // exp(-L2/(16 bw))
        const float e2  = e * e, e4 = e2 * e2, e8 = e4 * e4, e16 = e8 * e8;
        acc += ei * ej * w * (e + e2 + e4 + e8 + e16);
    }
    return acc;
}

// Main fused kernel: 128x64 macrotile per block (8 waves as 4x2), each wave does a
// 32x32 region = 2x2 tiles of 16x16 via V_WMMA_F32_16X16X4_F32 with 2 A-blocks x
// 2 B-blocks register blocking (4 loads : 4 WMMA per k-step). Software-pipelined:
// next k-step's operands are fetched before the current WMMA group executes.
// Upper-triangular macrotiles only; per-element triangular weights handle the
// diagonal band.
__global__ void __launch_bounds__(256) mmd_main_kernel(const float* __restrict__ src,
                                                       const float* __restrict__ tgt,
                                                       float* __restrict__ ws) {
    const int bI = blockIdx.y * 128;
    const int bJ = blockIdx.x * 64;
    if (bJ + 64 <= bI) return;             // entirely below diagonal (block-uniform)

    const float* __restrict__ sq = ws + 512;
    const float scale = 0.0625f / ws[2];   // 1/(16*bw)

    const int lane  = threadIdx.x & 31;
    const int wave  = threadIdx.x >> 5;
    const int m16   = lane & 15;   // row-within-tile for A/B operands; N index for C
    const int khalf = lane >> 4;   // which K pair this half-wave holds

    const int wi = wave >> 1;      // 0..3 : 32-row I sub-block
    const int wj = wave & 1;       // 0..1 : 32-row J sub-block

    const int rowA0 = bI + wi * 32 + m16;
    const int rowA1 = rowA0 + 16;
    const int rowB0 = bJ + wj * 32 + m16;
    const int rowB1 = rowB0 + 16;

    const float* pA0 = ((rowA0 < B_HALF) ? (src + (size_t)rowA0 * DIM)
                                         : (tgt + (size_t)(rowA0 - B_HALF) * DIM)) + 2 * khalf;
    const float* pA1 = ((rowA1 < B_HALF) ? (src + (size_t)rowA1 * DIM)
                                         : (tgt + (size_t)(rowA1 - B_HALF) * DIM)) + 2 * khalf;
    const float* pB0 = ((rowB0 < B_HALF) ? (src + (size_t)rowB0 * DIM)
                                         : (tgt + (size_t)(rowB0 - B_HALF) * DIM)) + 2 * khalf;
    const float* pB1 = ((rowB1 < B_HALF) ? (src + (size_t)rowB1 * DIM)
                                         : (tgt + (size_t)(rowB1 - B_HALF) * DIM)) + 2 * khalf;

    v8f c00 = {}, c01 = {}, c10 = {}, c11 = {};

    // prologue: fetch k = 0 operands
    v2f a0 = *(const v2f*)(pA0);
    v2f a1 = *(const v2f*)(pA1);
    v2f b0 = *(const v2f*)(pB0);
    v2f b1 = *(const v2f*)(pB1);

    #pragma unroll 4
    for (int k = 0; k < DIM - 4; k += 4) {
        // prefetch next k-step before consuming current operands
        v2f na0 = *(const v2f*)(pA0 + k + 4);
        v2f na1 = *(const v2f*)(pA1 + k + 4);
        v2f nb0 = *(const v2f*)(pB0 + k + 4);
        v2f nb1 = *(const v2f*)(pB1 + k + 4);

        c00 = __builtin_amdgcn_wmma_f32_16x16x4_f32(false, a0, false, b0, (short)0, c00, false, false);
        c01 = __builtin_amdgcn_wmma_f32_16x16x4_f32(false, a0, false, b1, (short)0, c01, false, false);
        c10 = __builtin_amdgcn_wmma_f32_16x16x4_f32(false, a1, false, b0, (short)0, c10, false, false);
        c11 = __builtin_amdgcn_wmma_f32_16x16x4_f32(false, a1, false, b1, (short)0, c11, false, false);

        a0 = na0; a1 = na1; b0 = nb0; b1 = nb1;
    }
    // epilogue k-step (k = DIM-4)
    c00 = __builtin_amdgcn_wmma_f32_16x16x4_f32(false, a0, false, b0, (short)0, c00, false, false);
    c01 = __builtin_amdgcn_wmma_f32_16x16x4_f32(false, a0, false, b1, (short)0, c01, false, false);
    c10 = __builtin_amdgcn_wmma_f32_16x16x4_f32(false, a1, false, b0, (short)0, c10, false, false);
    c11 = __builtin_amdgcn_wmma_f32_16x16x4_f32(false, a1, false, b1, (short)0, c11, false, false);

    // C/D layout: lane L, VGPR r -> m = r + 8*(L>>4), n = L & 15
    const int gj0 = bJ + wj * 32 + m16;
    const int gj1 = gj0 + 16;
    const int giB0 = bI + wi * 32 + 8 * khalf;
    const int giB1 = giB0 + 16;

    const float sqj0 = sq[gj0];
    const float sqj1 = sq[gj1];
    const float ej0 = (gj0 < B_HALF) ? 1.0f : -1.0f;
    const float ej1 = (gj1 < B_HALF) ? 1.0f : -1.0f;

    float local = 0.0f;
    local += tile_contrib(c00, giB0, gj0, sqj0, ej0, sq, scale);
    local += tile_contrib(c01, giB0, gj1, sqj1, ej1, sq, scale);
    local += tile_contrib(c10, giB1, gj0, sqj0, ej0, sq, scale);
    local += tile_contrib(c11, giB1, gj1, sqj1, ej1, sq, scale);

    __shared__ float red[256];
    red[threadIdx.x] = local;
    __syncthreads();
    for (int s = 128; s > 0; s >>= 1) {
        if (threadIdx.x < (unsigned)s) red[threadIdx.x] += red[threadIdx.x + s];
        __syncthreads();
    }
    if (threadIdx.x == 0) atomicAdd(&ws[1], red[0]);
}

__global__ void mmd_finalize_kernel(const float* __restrict__ ws, float* __restrict__ out) {
    // mean over b^2 elements, kernels averaged over 5 sigmas
    out[0] = ws[1] * (1.0f / (5.0f * (float)B_HALF * (float)B_HALF));
}

extern "C" void kernel_launch(void* const* d_in, const int* in_sizes, int n_in,
                              void* d_out, int out_size, void* d_ws, size_t ws_size,
                              hipStream_t stream) {
    const float* src = (const float*)d_in[0];
    const float* tgt = (const float*)d_in[1];
    float* ws  = (float*)d_ws;
    float* out = (float*)d_out;

    mmd_init_kernel<<<1, 256, 0, stream>>>(ws);
    mmd_rownorm_kernel<<<N_TOTAL, 256, 0, stream>>>(src, tgt, ws);
    mmd_colsum_kernel<<<N_TOTAL / 128, 256, 0, stream>>>(src, tgt, ws);
    mmd_bw_kernel<<<1, 256, 0, stream>>>(ws);

    dim3 grid(N_TOTAL / 64, N_TOTAL / 128);  // 128 x 64 blocks of 64(J) x 128(I)
    mmd_main_kernel<<<grid, 256, 0, stream>>>(src, tgt, ws);

    mmd_finalize_kernel<<<1, 1, 0, stream>>>(ws, out);
}